// HiPPO_43542378447441
// MI455X (gfx1250) — compile-verified
//
#include <hip/hip_runtime.h>

typedef float v2f __attribute__((ext_vector_type(2)));
typedef float v4f __attribute__((ext_vector_type(4)));
typedef float v8f __attribute__((ext_vector_type(8)));

#define SEQ   1024
#define HEADS 256
#define NST   64
#define COLS  16     // sequences (columns) per workgroup = WMMA N
#define CSTR  68     // padded column stride (floats): 64 + 4 -> conflict-free ds_load_b64

// 8 waves: wave = wm + 4*wk. wm selects 16-row block of x_new, wk selects K-half.
__launch_bounds__(256)
__global__ void hippo_scan_kernel(const float* __restrict__ inp,   // [B,S,H]
                                  const float* __restrict__ Amat,  // [N,N] row-major
                                  const float* __restrict__ Bvec,  // [N]
                                  const float* __restrict__ x0,    // [B,H,N]
                                  float* __restrict__ out)         // [B,S,H,N]
{
    __shared__ __attribute__((aligned(16))) float s_state[COLS * CSTR];   // transposed [col][row]
    __shared__ __attribute__((aligned(16))) float s_part[2][COLS * CSTR]; // per-K-half partials

    const int tid  = threadIdx.x;
    const int lane = tid & 31;
    const int wave = tid >> 5;
    const int wm   = wave & 3;      // row block: rows 16*wm .. 16*wm+15
    const int wk   = wave >> 2;     // K half:   k in [32*wk, 32*wk+32)
    const int half = lane >> 4;     // WMMA half-lane
    const int l16  = lane & 15;

    const int wg = blockIdx.x;      // 64 WGs: 4 batches x 16 head-groups
    const int b  = wg >> 4;
    const int h0 = (wg & 15) << 4;

    // ---- A operand resident in VGPRs for all 1024 steps ----
    // A-tile (16x4 f32) layout: lane(r=l16, h): VGPR0 = A[r][kb+2h], VGPR1 = A[r][kb+2h+1]
    v2f a[8];
    {
        const float* ap = Amat + (16 * wm + l16) * NST + (wk * 32 + 2 * half);
#pragma unroll
        for (int kt = 0; kt < 8; ++kt)
            a[kt] = *(const v2f*)(ap + 4 * kt);
    }
    // ---- B-vector entries for this lane's D rows; zeroed on wk=1 so B*u enters once ----
    float bvw[8];
    {
        const float* bp  = Bvec + 16 * wm + 8 * half;
        const float  msk = (wk == 0) ? 1.0f : 0.0f;
#pragma unroll
        for (int v = 0; v < 8; ++v) bvw[v] = msk * bp[v];
    }
    // ---- initial state -> LDS (transposed, padded) ----
    for (int i = tid; i < COLS * NST; i += 256) {
        const int c = i >> 6, m = i & 63;
        s_state[c * CSTR + m] = x0[((size_t)b * HEADS + h0 + c) * NST + m];
    }
    __syncthreads();

    const float* up  = inp + (size_t)b * SEQ * HEADS + h0 + l16;  // u[t, col=l16]
    const int pidx   = l16 * CSTR + 16 * wm + 8 * half;           // this lane's D slice
    const int ocol   = 2 * wave + half;                           // output copy column
    float* op = out + ((size_t)b * SEQ * HEADS + (size_t)(h0 + ocol)) * NST + l16 * 4;

    float*       mypart = &s_part[wk][0];
    const float* peer   = &s_part[wk ^ 1][0];

    float u = up[0];                                              // prefetched drive
    for (int t = 0; t < SEQ; ++t) {
        // state -> WMMA B-operand tiles: lane(n=l16,h): {X[4k+2h][n], X[4k+2h+1][n]}
        v2f x[8];
        const float* srow = s_state + l16 * CSTR + wk * 32 + 2 * half;
#pragma unroll
        for (int kt = 0; kt < 8; ++kt)
            x[kt] = *(const v2f*)(srow + 4 * kt);

        // prefetch next step's drive (clamped last iter; all waves load -> branch-free)
        const int   tn     = (t + 1 < SEQ) ? t + 1 : t;
        const float u_next = up[(size_t)tn * HEADS];

        // fold B*u into C-init (wk=1 waves contribute zero); two chains for ILP
        v8f acc0, acc1;
#pragma unroll
        for (int v = 0; v < 8; ++v) { acc0[v] = bvw[v] * u; acc1[v] = 0.0f; }

#pragma unroll
        for (int kt = 0; kt < 8; kt += 2) {
            acc0 = __builtin_amdgcn_wmma_f32_16x16x4_f32(
                false, a[kt],     false, x[kt],     (short)0, acc0, false, false);
            acc1 = __builtin_amdgcn_wmma_f32_16x16x4_f32(
                false, a[kt + 1], false, x[kt + 1], (short)0, acc1, false, false);
        }

        v4f lo, hi;
#pragma unroll
        for (int v = 0; v < 4; ++v) {
            lo[v] = acc0[v]     + acc1[v];
            hi[v] = acc0[v + 4] + acc1[v + 4];
        }

        // symmetric K-half exchange: publish own partial, read peer's
        *(v4f*)(mypart + pidx)     = lo;
        *(v4f*)(mypart + pidx + 4) = hi;
        __syncthreads();

        const v4f plo = *(const v4f*)(peer + pidx);
        const v4f phi = *(const v4f*)(peer + pidx + 4);
#pragma unroll
        for (int v = 0; v < 4; ++v) { lo[v] += plo[v]; hi[v] += phi[v]; }
        // both K-half waves commit identical bits (IEEE add commutes) -> benign duplicate store
        *(v4f*)&s_state[pidx]     = lo;
        *(v4f*)&s_state[pidx + 4] = hi;
        __syncthreads();

        // coalesced trajectory write: 2 columns/wave, 512B bursts, 4KB per WG per step
        const v4f val = *(const v4f*)&s_state[ocol * CSTR + l16 * 4];
        *(v4f*)(op + (size_t)t * (HEADS * NST)) = val;

        u = u_next;
    }
}

extern "C" void kernel_launch(void* const* d_in, const int* in_sizes, int n_in,
                              void* d_out, int out_size, void* d_ws, size_t ws_size,
                              hipStream_t stream) {
    (void)in_sizes; (void)n_in; (void)d_ws; (void)ws_size; (void)out_size;
    const float* inp  = (const float*)d_in[0];   // [4,1024,256]
    const float* Amat = (const float*)d_in[1];   // [64,64]
    const float* Bvec = (const float*)d_in[2];   // [64]
    const float* x0   = (const float*)d_in[3];   // [4,256,64]
    float* out = (float*)d_out;                  // [4,1024,256,64]
    hippo_scan_kernel<<<dim3(64), dim3(256), 0, stream>>>(inp, Amat, Bvec, x0, out);
}